// SSIMLoss_6863357739718
// MI455X (gfx1250) — compile-verified
//
#include <hip/hip_runtime.h>

// SSIM loss for MI455X (gfx1250, wave32).
// Horizontal Gaussian pass runs on the matrix pipe via V_WMMA_F32_16X16X4_F32
// (out_tile(16x16) = tmp(16x28) x BandedGauss(28x16), 7 K=4 steps).
// Vertical pass + products on VALU through LDS. One wave per 16x16 tile so
// EXEC is all ones at every WMMA (ISA 7.12 requirement).

typedef __attribute__((ext_vector_type(2))) float v2f;
typedef __attribute__((ext_vector_type(8))) float v8f;

#define IMG_W     512
#define IMG_H     512
#define N_IMG     16
#define TILE      16
#define PAD       5
#define WSZ       11
#define WIN       26          // TILE + WSZ - 1
#define IN_STRIDE 27          // WIN + 1 (bank-conflict pad)
#define TMP_STRIDE 28         // K padded to 28 (7 x K=4 WMMA steps)
#define NT        (N_IMG * 32 * 32)          // 16384 tiles
#define NPIX      4194304.0f                 // 16*1*512*512

// 11-tap Gaussian, sigma = 1.5, normalized.
__device__ const float GW[11] = {
    0.00102845f, 0.00759870f, 0.03600077f, 0.10936069f, 0.21300553f,
    0.26601172f, 0.21300553f, 0.10936069f, 0.03600077f, 0.00759870f,
    0.00102845f};

__global__ __launch_bounds__(32)
void ssim_tile_kernel(const float* __restrict__ fused,
                      const float* __restrict__ imga,
                      const float* __restrict__ imgb,
                      float* __restrict__ partial)
{
    __shared__ float sA[WIN * IN_STRIDE];
    __shared__ float sB[WIN * IN_STRIDE];
    __shared__ float sC[WIN * IN_STRIDE];
    __shared__ float Gs[16];
    __shared__ float tmp[8 * TILE * TMP_STRIDE];   // 8 conv channels

    const int lane = threadIdx.x;                  // 0..31 (one wave)
    const int tx = blockIdx.x, ty = blockIdx.y, n = blockIdx.z;
    const int gx0 = tx * TILE - PAD;
    const int gy0 = ty * TILE - PAD;

    const float* __restrict__ fb = fused + (size_t)n * (IMG_W * IMG_H);
    const float* __restrict__ ab = imga  + (size_t)n * (IMG_W * IMG_H);
    const float* __restrict__ bb = imgb  + (size_t)n * (IMG_W * IMG_H);

    if (lane < 16) Gs[lane] = (lane < WSZ) ? GW[lane] : 0.f;

    // ---- load 26x26 halo window, map [-1,1] -> [0,1], zero-pad OOB ----
    for (int t = 0; t < 22; ++t) {
        int e = lane + t * 32;
        if (e < WIN * WIN) {
            int wy = e / WIN;
            int wx = e - wy * WIN;
            int gy = gy0 + wy, gx = gx0 + wx;
            bool ok = (gy >= 0) && (gy < IMG_H) && (gx >= 0) && (gx < IMG_W);
            size_t idx = ok ? ((size_t)gy * IMG_W + gx) : 0;
            float a = ok ? (fb[idx] + 1.f) * 0.5f : 0.f;
            float b = ok ? (ab[idx] + 1.f) * 0.5f : 0.f;
            float c = ok ? (bb[idx] + 1.f) * 0.5f : 0.f;
            int o = wy * IN_STRIDE + wx;
            sA[o] = a; sB[o] = b; sC[o] = c;
        }
    }

    // zero K-pad columns 26,27 of every tmp channel (8 ch * 16 rows * 2 cols)
    {
        int row = lane & 15, col = WIN + (lane >> 4);
        #pragma unroll
        for (int cv = 0; cv < 8; ++cv)
            tmp[cv * (TILE * TMP_STRIDE) + row * TMP_STRIDE + col] = 0.f;
    }
    __syncthreads();

    // ---- vertical 11-tap pass for 8 channels: tmp(16 x 26) ----
    for (int t = 0; t < 13; ++t) {           // 13*32 == 16*26 exactly
        int e = lane + t * 32;
        int y = e / WIN;
        int x = e - y * WIN;
        int base = y * IN_STRIDE + x;
        float sa = 0.f, sb = 0.f, sc = 0.f;
        float saa = 0.f, sbb = 0.f, scc = 0.f, sab = 0.f, sac = 0.f;
        #pragma unroll
        for (int k = 0; k < WSZ; ++k) {
            float g = Gs[k];
            float a = sA[base + k * IN_STRIDE];
            float b = sB[base + k * IN_STRIDE];
            float c = sC[base + k * IN_STRIDE];
            sa  += g * a;        sb  += g * b;        sc  += g * c;
            saa += g * (a * a);  sbb += g * (b * b);  scc += g * (c * c);
            sab += g * (a * b);  sac += g * (a * c);
        }
        int o = y * TMP_STRIDE + x;
        tmp[0 * 448 + o] = sa;  tmp[1 * 448 + o] = sb;  tmp[2 * 448 + o] = sc;
        tmp[3 * 448 + o] = saa; tmp[4 * 448 + o] = sbb; tmp[5 * 448 + o] = scc;
        tmp[6 * 448 + o] = sab; tmp[7 * 448 + o] = sac;
    }
    __syncthreads();

    // ---- horizontal pass as WMMA: D = A(16x28) x B(28x16), 7 K=4 chunks ----
    // B fragment layout (f32 4x16): lanes 0-15 -> N=lane, v0=K0,v1=K1;
    //                               lanes 16-31 -> N=lane-16, v0=K2,v1=K3.
    const int half = lane >> 4;
    const int nn   = lane & 15;
    const int sub  = half * 2;
    v2f bfrag[7];
    #pragma unroll
    for (int kc = 0; kc < 7; ++kc) {
        int j0 = 4 * kc + sub;
        int d0 = j0 - nn, d1 = j0 + 1 - nn;        // banded Gauss: B[j][n]=g[j-n]
        bfrag[kc].x = (d0 >= 0 && d0 < WSZ) ? Gs[d0] : 0.f;
        bfrag[kc].y = (d1 >= 0 && d1 < WSZ) ? Gs[d1] : 0.f;
    }

    // A fragment (f32 16x4): lanes 0-15 M=lane (K0/K1), lanes 16-31 M=lane-16 (K2/K3)
    const int row = lane & 15;
    v8f acc[8];
    #pragma unroll
    for (int cv = 0; cv < 8; ++cv) {
        v8f c = {0.f, 0.f, 0.f, 0.f, 0.f, 0.f, 0.f, 0.f};
        #pragma unroll
        for (int kc = 0; kc < 7; ++kc) {
            int o = cv * 448 + row * TMP_STRIDE + 4 * kc + sub;
            v2f a;
            a.x = tmp[o];
            a.y = tmp[o + 1];
            c = __builtin_amdgcn_wmma_f32_16x16x4_f32(
                    /*neg_a=*/false, a, /*neg_b=*/false, bfrag[kc],
                    /*c_mod=*/(short)0, c, /*reuse_a=*/false, /*reuse_b=*/false);
        }
        acc[cv] = c;
    }

    // ---- per-pixel SSIM combine (8 pixels per lane) ----
    const float C1v = 0.0001f, C2v = 0.0009f;
    float sumA = 0.f, sumB = 0.f;
    #pragma unroll
    for (int i = 0; i < 8; ++i) {
        float m1  = acc[0][i];   // mu fused
        float m2  = acc[1][i];   // mu img_a
        float m3  = acc[2][i];   // mu img_b
        float eaa = acc[3][i], ebb = acc[4][i], ecc = acc[5][i];
        float eab = acc[6][i], eac = acc[7][i];
        {
            float s1 = eaa - m1 * m1, s2 = ebb - m2 * m2, s12 = eab - m1 * m2;
            float num = (2.f * m1 * m2 + C1v) * (2.f * s12 + C2v);
            float den = (m1 * m1 + m2 * m2 + C1v) * (s1 + s2 + C2v);
            sumA += num / den;
        }
        {
            float s1 = eaa - m1 * m1, s2 = ecc - m3 * m3, s12 = eac - m1 * m3;
            float num = (2.f * m1 * m3 + C1v) * (2.f * s12 + C2v);
            float den = (m1 * m1 + m3 * m3 + C1v) * (s1 + s2 + C2v);
            sumB += num / den;
        }
    }

    // ---- wave32 reduction, one deterministic partial per tile ----
    #pragma unroll
    for (int off = 16; off > 0; off >>= 1) {
        sumA += __shfl_xor(sumA, off, 32);
        sumB += __shfl_xor(sumB, off, 32);
    }
    if (lane == 0) {
        int tid = (n * 32 + ty) * 32 + tx;
        partial[tid]      = sumA;
        partial[NT + tid] = sumB;
    }
}

__global__ __launch_bounds__(256)
void ssim_reduce_kernel(const float* __restrict__ partial, float* __restrict__ out)
{
    __shared__ float rA[256];
    __shared__ float rB[256];
    int t = threadIdx.x;
    float sA = 0.f, sB = 0.f;
    for (int i = t; i < NT; i += 256) {
        sA += partial[i];
        sB += partial[NT + i];
    }
    rA[t] = sA; rB[t] = sB;
    __syncthreads();
    for (int s = 128; s > 0; s >>= 1) {
        if (t < s) { rA[t] += rA[t + s]; rB[t] += rB[t + s]; }
        __syncthreads();
    }
    if (t == 0) {
        float ma = rA[0] / NPIX;
        float mb = rB[0] / NPIX;
        out[0] = 1.f - 0.5f * (ma + mb);
    }
}

extern "C" void kernel_launch(void* const* d_in, const int* in_sizes, int n_in,
                              void* d_out, int out_size, void* d_ws, size_t ws_size,
                              hipStream_t stream) {
    (void)in_sizes; (void)n_in; (void)out_size; (void)ws_size;
    const float* fused = (const float*)d_in[0];
    const float* imga  = (const float*)d_in[1];
    const float* imgb  = (const float*)d_in[2];
    float* partial = (float*)d_ws;          // 2*NT floats = 128 KB
    float* out = (float*)d_out;

    dim3 grid(32, 32, N_IMG);               // one wave per 16x16 tile
    ssim_tile_kernel<<<grid, 32, 0, stream>>>(fused, imga, imgb, partial);
    ssim_reduce_kernel<<<1, 256, 0, stream>>>(partial, out);
}